// KnowledgeStore_12506944766451
// MI455X (gfx1250) — compile-verified
//
#include <hip/hip_runtime.h>
#include <hip/hip_bf16.h>
#include <float.h>

typedef float v2f __attribute__((ext_vector_type(2)));
typedef float v8f __attribute__((ext_vector_type(8)));

#define KTOT   8192
#define WN     32
#define TD     512
#define FN     12
#define VD     512
#define BATCH  8

#define DT     16              // d-columns per block (== waves per block)
#define KT     16              // k rows per WMMA tile (M)
#define KSPLIT 32              // k-dimension split across blocks
#define KPB    (KTOT / KSPLIT) // 256 k per block
#define NTILES (KPB / KT)      // 16 tiles per block
#define TILE_F (KT * WN * DT)  // floats per staged tile (8192 = 32 KB)

// ---------------------------------------------------------------------------
// Kernel 1: lq2s[b][d][w] = log2(query[b][w][d])   (B operand table, 512 KB)
// ---------------------------------------------------------------------------
__global__ void prelog_kernel(const float* __restrict__ query,
                              float* __restrict__ lq2s) {
    int o = blockIdx.x * blockDim.x + threadIdx.x;  // 0 .. 131071
    int b = o >> 14;           // / (TD*WN)
    int r = o & 16383;
    int d = r >> 5;
    int w = r & 31;
    lq2s[o] = __log2f(query[(b * WN + w) * TD + d]);
}

// ---------------------------------------------------------------------------
// Kernel 2: per (b,d), partial argmin over this block's k-range via WMMA.
//   kl2[b,k,d] = sum_w qt*log2(qt)  -  sum_w qt*log2(query[b])   (scale-inv.)
// Staging: double-buffered GLOBAL_LOAD_ASYNC_TO_LDS_B32 (ASYNCcnt), per-lane
// LDS addresses carry the d^k XOR swizzle -> conflict-free WMMA A reads.
// Self term computed in-wave and folded into the WMMA accumulator (C = -s),
// so the argmin tail is register-only: argmin(s-cross) == argmax(cross-s).
// ---------------------------------------------------------------------------
__global__ __launch_bounds__(512)
void kl_argmin_kernel(const float* __restrict__ qt,
                      const float* __restrict__ lq2s,
                      float* __restrict__ pmin,
                      int*   __restrict__ pidx) {
    __shared__ float qts[2][TILE_F];      // XOR-swizzled [k][w][d^k], x2 buffers

    const int tid  = threadIdx.x;
    const int db   = blockIdx.x % (TD / DT);   // 32 d-blocks
    const int kb   = blockIdx.x / (TD / DT);   // 32 k-blocks
    const int d0   = db * DT;
    const int kbas = kb * KPB;

    const int lane = tid & 31;
    const int wave = tid >> 5;        // 0..15 -> this wave's local d
    const int h    = lane >> 4;       // half: selects K-inner pair {0,1} vs {2,3}
    const int n    = lane & 15;       // N (= b slot) for B/C; M row for A/self reads

    const int dl = wave;              // local d of this wave
    const int d  = d0 + dl;           // global d

    const int dstage = tid & 15;      // staging: lanes across d (coalesced)
    const int wstage = tid >> 4;      // 0..31

    // LDS base byte-offset of qts (generic LDS pointer: low 32 bits = offset)
    const uint32_t lds0  = (uint32_t)(uintptr_t)(&qts[0][0]);
    const uint32_t gbase = (uint32_t)wstage * (TD * 4u) + (uint32_t)(d0 + dstage) * 4u;

    // B operands: 4x16 f32 tiles, VGPR0 = K=2h, VGPR1 = K=2h+1, cols = b.
    // k-invariant -> load once, keep in registers. Cols 8..15 are zero.
    v2f Bop[8];
#pragma unroll
    for (int j = 0; j < 8; ++j) { Bop[j].x = 0.0f; Bop[j].y = 0.0f; }
    if (n < BATCH) {
        const float* base = lq2s + ((size_t)n * TD + d) * WN;
#pragma unroll
        for (int j = 0; j < 8; ++j) {
            int w = 4 * j + 2 * h;
            Bop[j].x = base[w];
            Bop[j].y = base[w + 1];
        }
    }

    float maxV = -FLT_MAX;            // running max of (cross - s) == -kl
    int   maxI = 0;

    // ---- prologue: async-stage tile 0 into buffer 0 ----
#pragma unroll
    for (int p = 0; p < KT; ++p) {
        uint32_t goff = (uint32_t)(kbas + p) * (WN * TD * 4u) + gbase;
        uint32_t loff = lds0 + (uint32_t)(p * (WN * DT) + wstage * DT + (dstage ^ p)) * 4u;
        asm volatile("global_load_async_to_lds_b32 %0, %1, %2"
                     :: "v"(loff), "v"(goff), "s"(qt) : "memory");
    }
    asm volatile("s_wait_asynccnt 0x0" ::: "memory");
    __syncthreads();

    for (int t = 0; t < NTILES; ++t) {
        const int cur = t & 1;
        const int k0  = kbas + t * KT;

        // ---- async-stage tile t+1 into the other buffer (overlaps compute) ----
        if (t + 1 < NTILES) {
            const uint32_t lbase = lds0 + (uint32_t)((cur ^ 1) * (TILE_F * 4));
#pragma unroll
            for (int p = 0; p < KT; ++p) {
                uint32_t goff = (uint32_t)(k0 + KT + p) * (WN * TD * 4u) + gbase;
                uint32_t loff = lbase + (uint32_t)(p * (WN * DT) + wstage * DT + (dstage ^ p)) * 4u;
                asm volatile("global_load_async_to_lds_b32 %0, %1, %2"
                             :: "v"(loff), "v"(goff), "s"(qt) : "memory");
            }
        }

        // ---- in-wave self term: lane (h,n) sums w in [16h,16h+16) for row n ----
        float part = 0.0f;
#pragma unroll
        for (int w = 0; w < 16; ++w) {
            float v = qts[cur][n * (WN * DT) + (16 * h + w) * DT + (dl ^ n)];
            part = fmaf(v, __log2f(v), part);
        }
        float sfull = part + __shfl_xor(part, 16, 32);  // s[kk=n][d], all 32 w

        // ---- WMMA accumulator seeded with -s (C VGPR i: rows i / i+8) ----
        v8f c;
#pragma unroll
        for (int i = 0; i < 8; ++i)
            c[i] = -__shfl(sfull, i + 8 * h, 32);

        // ---- cross term via 8x V_WMMA_F32_16X16X4_F32 (uniform flow) ----
#pragma unroll
        for (int j = 0; j < 8; ++j) {
            int w = 4 * j + 2 * h;            // K-inner origin for this half
            v2f a;                            // A 16x4: lane holds (M=n, w), (M=n, w+1)
            a.x = qts[cur][n * (WN * DT) + w * DT + (dl ^ n)];
            a.y = qts[cur][n * (WN * DT) + (w + 1) * DT + (dl ^ n)];
            c = __builtin_amdgcn_wmma_f32_16x16x4_f32(
                    false, a, false, Bop[j], (short)0, c, false, false);
        }

        // ---- register-only argmax of (cross - s); k ascending -> first ties ----
#pragma unroll
        for (int i = 0; i < 8; ++i) {
            int kg = k0 + i + 8 * h;          // C VGPR i: rows i (lo) / i+8 (hi)
            if (c[i] > maxV) { maxV = c[i]; maxI = kg; }
        }

        // ---- next buffer resident before anyone reads it; cur reusable ----
        asm volatile("s_wait_asynccnt 0x0" ::: "memory");
        __syncthreads();
    }

    // combine lane halves (lo holds k%16<8, hi holds k%16>=8 candidates)
    float hv = __shfl(maxV, n + 16, 32);
    int   hi = __shfl(maxI, n + 16, 32);
    if (h == 0 && n < BATCH) {
        if (hv > maxV || (hv == maxV && hi < maxI)) { maxV = hv; maxI = hi; }
        pmin[(kb * BATCH + n) * TD + d] = -maxV;   // back to kl-min convention
        pidx[(kb * BATCH + n) * TD + d] = maxI;
    }
}

// ---------------------------------------------------------------------------
// Kernel 3: reduce KSPLIT partials per (b,d) and gather queue_video rows.
// ---------------------------------------------------------------------------
__global__ void gather_kernel(const float* __restrict__ pmin,
                              const int*   __restrict__ pidx,
                              const float* __restrict__ qv,
                              float* __restrict__ out) {
    const int bd = blockIdx.x;            // 0 .. B*TD-1
    const int b  = bd / TD;
    const int d  = bd % TD;

    float best = FLT_MAX;
    int   bi   = 0;
    for (int kb = 0; kb < KSPLIT; ++kb) {           // ascending k ranges
        float v = pmin[(kb * BATCH + b) * TD + d];
        int   i = pidx[(kb * BATCH + b) * TD + d];
        if (v < best || (v == best && i < bi)) { best = v; bi = i; }
    }

    const float4* src = (const float4*)(qv + (size_t)bi * (FN * VD));
    float4*       dst = (float4*)(out + (size_t)bd * (FN * VD));
    for (int i = threadIdx.x; i < (FN * VD) / 4; i += blockDim.x)
        dst[i] = src[i];
}

// ---------------------------------------------------------------------------
extern "C" void kernel_launch(void* const* d_in, const int* in_sizes, int n_in,
                              void* d_out, int out_size, void* d_ws, size_t ws_size,
                              hipStream_t stream) {
    (void)in_sizes; (void)n_in; (void)out_size; (void)ws_size;
    const float* query       = (const float*)d_in[0];  // [B, W, TD]
    const float* queue_text  = (const float*)d_in[1];  // [K, W, TD]
    const float* queue_video = (const float*)d_in[2];  // [K, FN, VD]
    float* out = (float*)d_out;                        // [B, TD, FN, VD]

    float* lq2s = (float*)d_ws;                          // B*TD*WN floats
    float* pmin = lq2s + (size_t)BATCH * TD * WN;        // KSPLIT*B*TD floats
    int*   pidx = (int*)(pmin + (size_t)KSPLIT * BATCH * TD);

    prelog_kernel<<<(BATCH * TD * WN) / 256, 256, 0, stream>>>(query, lq2s);
    kl_argmin_kernel<<<(TD / DT) * KSPLIT, 512, 0, stream>>>(queue_text, lq2s, pmin, pidx);
    gather_kernel<<<BATCH * TD, 256, 0, stream>>>(pmin, pidx, queue_video, out);
}